// TorchRecurrentPolicy_73521250173176
// MI455X (gfx1250) — compile-verified
//
#include <hip/hip_runtime.h>
#include <hip/hip_bf16.h>
#include <math.h>

typedef __attribute__((ext_vector_type(2))) float v2f;
typedef __attribute__((ext_vector_type(8))) float v8f;
typedef int v4i __attribute__((vector_size(16)));
typedef __attribute__((address_space(1))) v4i* gv4i_p;   // global int4*
typedef __attribute__((address_space(3))) v4i* lv4i_p;   // LDS int4*

#define BB    2048            // batch
#define FF    512             // obs features
#define NN    2048            // nodes
#define NOUT  64              // output keys
#define EIN   32768
#define EREC  32768
#define KTOT  (FF + NN)       // 2560 = concat [obs | prev_state] feature axis
#define NSEG  4               // split-K segments
#define KSEG  (KTOT / NSEG)   // 640
#define KC    64              // K chunk per LDS stage (obs/prev boundary is a multiple of 64)
#define KCP   68              // padded LDS row stride: 16B-aligned rows + conflict-free frags
#define NB    32              // batch columns per workgroup (2 N-tiles x 4 M-tiles = 8 waves)

// ---------------------------------------------------------- async global->LDS
#if __has_builtin(__builtin_amdgcn_global_load_async_to_lds_b128)
#define HAVE_ASYNC 1
#else
#define HAVE_ASYNC 0
#endif

__device__ __forceinline__ void copy16_to_lds(const float* g, float* l) {
#if HAVE_ASYNC
  // ASYNCcnt-tracked DMA-style copy, bypasses VGPRs (CDNA5 ISA).
  __builtin_amdgcn_global_load_async_to_lds_b128(
      (gv4i_p)(void*)g, (lv4i_p)(void*)l, 0, 0);
#else
  *(float4*)l = *(const float4*)g;
#endif
}

__device__ __forceinline__ void wait_stage_barrier() {
#if HAVE_ASYNC
  asm volatile("s_wait_asynccnt 0x0" ::: "memory");
#endif
  __syncthreads();
}

// ---------------------------------------------------------------- zero scratch
__global__ void zero_ws_kernel(float* __restrict__ Wd, int n) {
  int i = blockIdx.x * blockDim.x + threadIdx.x;
  if (i < n) Wd[i] = 0.0f;
}

// ------------------------------------------------- scatter edges -> dense rows
// Wd is [NOUT x KTOT]: row k holds all weights feeding output_keys[k].
// atomicAdd accumulates duplicate (src,dst) edges; duplicate keys get own rows.
__global__ __launch_bounds__(256) void densify_kernel(
    const float* __restrict__ w_in,   const float* __restrict__ w_rec,
    const int*   __restrict__ src_in,  const int* __restrict__ dst_in,
    const int*   __restrict__ src_rec, const int* __restrict__ dst_rec,
    const int*   __restrict__ okeys,   float* __restrict__ Wd) {
  __shared__ int sk[NOUT];
  int tid = threadIdx.x;
  if (tid < NOUT) sk[tid] = okeys[tid];
  __syncthreads();

  int e = blockIdx.x * blockDim.x + tid;   // 0 .. EIN+EREC-1
  int d, col; float w;
  if (e < EIN) {
    d = dst_in[e];   col = src_in[e];       w = w_in[e];
  } else {
    int e2 = e - EIN;
    d = dst_rec[e2]; col = FF + src_rec[e2]; w = w_rec[e2];
  }
  for (int k = 0; k < NOUT; ++k)
    if (sk[k] == d) atomicAdd(&Wd[k * KTOT + col], w);
}

// ---------------------------------------------- split-K WMMA fp32 GEMM partial
// part[seg, b, k] = sum_{K in seg} Wd[k,K] * X[K,b]
//   X[K,b] = obs[b,K] (K < FF) else prev[b, K-FF]
// Grid: (BB/NB batch blocks) x (NSEG K-segments). Block: 8 waves,
// wave w -> m_base=(w&3)*16, n_loc=(w>>2)*16. Double-buffered async staging.
__global__ __launch_bounds__(256) void gemm_partial_kernel(
    const float* __restrict__ obs, const float* __restrict__ prev,
    const float* __restrict__ Wd,  float* __restrict__ part) {
  __shared__ float Wl[2][64 * KCP];   // [buf][m][k]
  __shared__ float Xl[2][NB * KCP];   // [buf][b][k]

  const int tid    = threadIdx.x;
  const int lane   = tid & 31;
  const int wv     = tid >> 5;
  const int lo     = lane & 15;       // row/col within half-wave
  const int hi     = lane >> 4;       // K-pair select for A/B fragments
  const int m_base = (wv & 3) * 16;
  const int n_loc  = (wv >> 2) * 16;
  const int b_base = blockIdx.x * NB;
  const int Kbeg   = blockIdx.y * KSEG;

  auto stage = [&](int buf, int K0) {
    // W chunk: 64 x KC floats = 1024 float4, 4 per thread, coalesced along K.
    for (int i = tid; i < 64 * (KC / 4); i += 256) {
      int r = i >> 4, c4 = (i & 15) * 4;
      copy16_to_lds(Wd + (size_t)r * KTOT + K0 + c4, &Wl[buf][r * KCP + c4]);
    }
    // X chunk: NB x KC floats = 512 float4, 2 per thread.
    const float* xsrc; int xs;
    if (K0 < FF) { xsrc = obs  + (size_t)b_base * FF + K0;        xs = FF; }
    else         { xsrc = prev + (size_t)b_base * NN + (K0 - FF); xs = NN; }
    for (int i = tid; i < NB * (KC / 4); i += 256) {
      int r = i >> 4, c4 = (i & 15) * 4;
      copy16_to_lds(xsrc + (size_t)r * xs + c4, &Xl[buf][r * KCP + c4]);
    }
  };

  stage(0, Kbeg);
  v8f acc = {};
  const int nch = KSEG / KC;          // 10 chunks per segment
  for (int c = 0; c < nch; ++c) {
    const int buf = c & 1;
    // Chunk c resident in LDS everywhere; chunk c-1 consumers are past this
    // barrier, so buf^1 is safe to overwrite with chunk c+1.
    wait_stage_barrier();
    if (c + 1 < nch) stage(buf ^ 1, Kbeg + (c + 1) * KC);

    // 16 fp32 WMMAs per chunk. A: 16x4 (lane=M, hi selects K pair);
    // B: 4x16 (lane=N/batch, hi selects K pair). EXEC all-ones here.
#pragma unroll
    for (int kk = 0; kk < KC; kk += 4) {
      v2f af = *(const v2f*)&Wl[buf][(m_base + lo) * KCP + kk + 2 * hi];
      v2f bf = *(const v2f*)&Xl[buf][(n_loc  + lo) * KCP + kk + 2 * hi];
      acc = __builtin_amdgcn_wmma_f32_16x16x4_f32(
          /*neg_a=*/false, af, /*neg_b=*/false, bf,
          /*c_mod=*/(short)0, acc, /*reuse_a=*/false, /*reuse_b=*/false);
    }
  }

  // D layout: VGPR j -> M = m_base + 8*hi + j, N = n_loc + lo (ISA 7.12.2).
  // 8 consecutive output slots per lane -> two float4 stores into partials.
  const int brow  = b_base + n_loc + lo;
  const int slot0 = m_base + 8 * hi;
  float* pp = part + (size_t)blockIdx.y * BB * NOUT + (size_t)brow * NOUT + slot0;
  *(float4*)(pp + 0) = make_float4(acc[0], acc[1], acc[2], acc[3]);
  *(float4*)(pp + 4) = make_float4(acc[4], acc[5], acc[6], acc[7]);
}

// -------------------------------------- fixed-order reduce + bias + tanh (det)
__global__ void bias_tanh_kernel(const float* __restrict__ part,
                                 const float* __restrict__ bias,
                                 const int* __restrict__ okeys,
                                 float* __restrict__ out) {
  int i = blockIdx.x * blockDim.x + threadIdx.x;   // over BB*NOUT
  if (i >= BB * NOUT) return;
  int k = i & (NOUT - 1);
  float s = bias[okeys[k]];
#pragma unroll
  for (int sgm = 0; sgm < NSEG; ++sgm)
    s += part[(size_t)sgm * BB * NOUT + i];        // fixed order -> deterministic
  out[i] = tanhf(s);
}

// ----------------------------------------------------------------------- entry
extern "C" void kernel_launch(void* const* d_in, const int* in_sizes, int n_in,
                              void* d_out, int out_size, void* d_ws, size_t ws_size,
                              hipStream_t stream) {
  const float* obs     = (const float*)d_in[0];
  const float* prev    = (const float*)d_in[1];
  const float* w_in    = (const float*)d_in[2];
  const float* w_rec   = (const float*)d_in[3];
  const float* bias    = (const float*)d_in[4];
  const int*   src_in  = (const int*)d_in[5];
  const int*   dst_in  = (const int*)d_in[6];
  const int*   src_rec = (const int*)d_in[7];
  const int*   dst_rec = (const int*)d_in[8];
  const int*   okeys   = (const int*)d_in[9];
  float* out  = (float*)d_out;
  float* Wd   = (float*)d_ws;                       // [NOUT x KTOT] = 640 KB
  float* part = Wd + (size_t)NOUT * KTOT;           // [NSEG x BB x NOUT] = 2 MB

  const int nW = NOUT * KTOT;
  zero_ws_kernel<<<(nW + 255) / 256, 256, 0, stream>>>(Wd, nW);
  densify_kernel<<<(EIN + EREC) / 256, 256, 0, stream>>>(
      w_in, w_rec, src_in, dst_in, src_rec, dst_rec, okeys, Wd);
  gemm_partial_kernel<<<dim3(BB / NB, NSEG), 256, 0, stream>>>(obs, prev, Wd, part);
  bias_tanh_kernel<<<(BB * NOUT + 255) / 256, 256, 0, stream>>>(part, bias, okeys, out);
}